// development_layer_62242666054211
// MI455X (gfx1250) — compile-verified
//
#include <hip/hip_runtime.h>

// CDNA5 (gfx1250) implementation of the unitary "development layer".
// Complex 8x8 algebra is embedded into real 16x16 ([[Re,-Im],[Im,Re]]),
// so every complex matmul is a single 16x16x16 f32 WMMA (4x v_wmma_f32_16x16x4_f32).

typedef __attribute__((ext_vector_type(2))) float v2f;
typedef __attribute__((ext_vector_type(8))) float v8f;

#define NN 64
#define TT 2048
#define DD 8
#define TM1 2047
#define NCHUNK 32
#define CHUNK 64   // NCHUNK*CHUNK = 2048 >= TM1 (tail padded with identity)

__device__ __forceinline__ void lds_fence() {
    // DS ops are in-order within a wave; this stops compiler reordering and
    // drains the LDS pipe so cross-lane store->load relayouts are safe.
    asm volatile("s_wait_dscnt 0" ::: "memory");
}

// Full 16x16x16 f32 matmul-accumulate: 4 K-slices of v_wmma_f32_16x16x4_f32.
// a[8]: A-matrix regs (A-layout), b[8]: B-matrix regs (B-layout), c: C/D-layout.
__device__ __forceinline__ v8f wmma_16x16x16(const float (&a)[8], const float (&b)[8], v8f c) {
#pragma unroll
    for (int s = 0; s < 4; ++s) {
        v2f av = { a[2 * s], a[2 * s + 1] };
        v2f bv = { b[2 * s], b[2 * s + 1] };
        c = __builtin_amdgcn_wmma_f32_16x16x4_f32(false, av, false, bv, (short)0, c,
                                                  false, false);
    }
    return c;
}

// Phase 1: each wave computes the ordered product of expm(AX_t) over one chunk.
__global__ __launch_bounds__(256) void dev_phase1(const float* __restrict__ x,
                                                  const float* __restrict__ A_re,
                                                  const float* __restrict__ A_im,
                                                  float* __restrict__ part) {
    __shared__ float sG[DD][16][16];   // 16x16 real embeddings of A_d - A_d^H
    __shared__ float sW[8][256];       // per-wave relayout scratch

    const int tid  = threadIdx.x;
    const int lane = tid & 31;
    const int wv   = tid >> 5;

    // Build generator embeddings: A = (Are + i Aim) - (Are + i Aim)^H
    //   Re = Are - Are^T, Im = Aim + Aim^T ; G = [[Re,-Im],[Im,Re]]
    for (int e = tid; e < DD * 256; e += 256) {
        int d = e >> 8, rc = e & 255, r = rc >> 4, c = rc & 15;
        int i = r & 7, j = c & 7;
        float re = A_re[d * 64 + i * 8 + j] - A_re[d * 64 + j * 8 + i];
        float im = A_im[d * 64 + i * 8 + j] + A_im[d * 64 + j * 8 + i];
        float v;
        if (r < 8) v = (c < 8) ? re : -im;
        else       v = (c < 8) ? im : re;
        sG[d][r][c] = v;
    }
    __syncthreads();

    const int gw    = blockIdx.x * 8 + wv;   // 0..2047
    const int n     = gw >> 5;               // sequence
    const int chunk = gw & (NCHUNK - 1);

    const int M = lane & 15;      // A-layout row / A-load row
    const int H = lane >> 4;      // lane half
    const int Ncol = M;           // B/C layouts: N = lane & 15
    int Kv[8], Mv[8];
#pragma unroll
    for (int v = 0; v < 8; ++v) {
        Kv[v] = 4 * (v >> 1) + (v & 1) + 2 * H;  // K index held by VGPR v (A/B layouts)
        Mv[v] = v + 8 * H;                       // row held by VGPR v (C/D layout)
    }
    float* w = sW[wv];

    // Running chunk product P, kept in A-layout. Init = Identity.
    float pa[8];
#pragma unroll
    for (int v = 0; v < 8; ++v) pa[v] = (M == Kv[v]) ? 1.0f : 0.0f;

    const float sscale = 1.0f / 32.0f;  // scaling-and-squaring: s = 5
    const long  xbase  = (long)n * (TT * DD);

    for (int tl = 0; tl < CHUNK; ++tl) {
        const int tg = chunk * CHUNK + tl;   // wave-uniform
        float dx[8];
        if (tg < TM1) {
            const float* xp = x + xbase + (long)tg * DD;
#pragma unroll
            for (int k = 0; k < 8; ++k) dx[k] = xp[DD + k] - xp[k];
        } else {
#pragma unroll
            for (int k = 0; k < 8; ++k) dx[k] = 0.0f;  // pad -> expm(0)=I
        }

        // Xs = (1/32) * sum_d dx[d] * G_d, built directly in A-layout.
        float xa[8];
#pragma unroll
        for (int v = 0; v < 8; ++v) {
            float s = 0.0f;
#pragma unroll
            for (int d = 0; d < 8; ++d) s += dx[d] * sG[d][M][Kv[v]];
            xa[v] = s * sscale;
        }

        // Taylor-Horner order 6: R <- I + (Xs * R) / j,  j = 6..1. R in B-layout.
        float bR[8];
#pragma unroll
        for (int v = 0; v < 8; ++v) bR[v] = (Kv[v] == Ncol) ? 1.0f : 0.0f;
#pragma unroll
        for (int j = 6; j >= 1; --j) {
            v8f dacc = {};
            dacc = wmma_16x16x16(xa, bR, dacc);
            const float inv = 1.0f / (float)j;
            lds_fence();  // prior loads of w done before overwrite
#pragma unroll
            for (int v = 0; v < 8; ++v)
                w[Mv[v] * 16 + Ncol] = dacc[v] * inv + ((Ncol == Mv[v]) ? 1.0f : 0.0f);
            lds_fence();
#pragma unroll
            for (int v = 0; v < 8; ++v) bR[v] = w[Kv[v] * 16 + Ncol];
        }

        // 5 squarings: E <- E*E  (E row-major lives in w; bR already holds E in B-layout)
#pragma unroll
        for (int sq = 0; sq < 5; ++sq) {
            float ea[8];
#pragma unroll
            for (int v = 0; v < 8; ++v) ea[v] = w[M * 16 + Kv[v]];  // A-layout of E
            v8f dacc = {};
            dacc = wmma_16x16x16(ea, bR, dacc);
            lds_fence();
#pragma unroll
            for (int v = 0; v < 8; ++v) w[Mv[v] * 16 + Ncol] = dacc[v];
            lds_fence();
#pragma unroll
            for (int v = 0; v < 8; ++v) bR[v] = w[Kv[v] * 16 + Ncol];
        }

        // P <- P @ E   (left-to-right time order)
        v8f dacc = {};
        dacc = wmma_16x16x16(pa, bR, dacc);
        lds_fence();
#pragma unroll
        for (int v = 0; v < 8; ++v) w[Mv[v] * 16 + Ncol] = dacc[v];
        lds_fence();
#pragma unroll
        for (int v = 0; v < 8; ++v) pa[v] = w[M * 16 + Kv[v]];
        // w now holds P row-major (reused/overwritten next step)
    }

    lds_fence();
    // Store chunk partial row-major: part[n][chunk][16][16]
    float* outp = part + (long)gw * 256;
#pragma unroll
    for (int q = 0; q < 8; ++q) outp[lane * 8 + q] = w[lane * 8 + q];
}

// Phase 2: one wave per n combines 32 chunk partials in order, emits complex64 output.
__global__ __launch_bounds__(256) void dev_phase2(const float* __restrict__ part,
                                                  float* __restrict__ out) {
    __shared__ float sW[8][256];
    __shared__ float sE[8][256];
    const int lane = threadIdx.x & 31;
    const int wv   = threadIdx.x >> 5;
    const int n    = blockIdx.x * 8 + wv;   // grid = 8 blocks -> 64 waves

    const int M = lane & 15;
    const int H = lane >> 4;
    const int Ncol = M;
    int Kv[8], Mv[8];
#pragma unroll
    for (int v = 0; v < 8; ++v) {
        Kv[v] = 4 * (v >> 1) + (v & 1) + 2 * H;
        Mv[v] = v + 8 * H;
    }
    float* w = sW[wv];
    float* e = sE[wv];

    const float* p0 = part + (long)n * NCHUNK * 256;
#pragma unroll
    for (int q = 0; q < 8; ++q) w[lane * 8 + q] = p0[lane * 8 + q];
    lds_fence();
    float pa[8];
#pragma unroll
    for (int v = 0; v < 8; ++v) pa[v] = w[M * 16 + Kv[v]];

    for (int c = 1; c < NCHUNK; ++c) {
        const float* pc = p0 + (long)c * 256;
#pragma unroll
        for (int q = 0; q < 8; ++q) e[lane * 8 + q] = pc[lane * 8 + q];
        lds_fence();
        float bR[8];
#pragma unroll
        for (int v = 0; v < 8; ++v) bR[v] = e[Kv[v] * 16 + Ncol];
        v8f dacc = {};
        dacc = wmma_16x16x16(pa, bR, dacc);
        lds_fence();
#pragma unroll
        for (int v = 0; v < 8; ++v) w[Mv[v] * 16 + Ncol] = dacc[v];
        lds_fence();
#pragma unroll
        for (int v = 0; v < 8; ++v) pa[v] = w[M * 16 + Kv[v]];
    }
    lds_fence();

    // Extract complex64: Re = P[i][j], Im = P[i+8][j]; interleaved output (N,1,8,8) c64.
#pragma unroll
    for (int q = 0; q < 2; ++q) {
        int idx = lane * 2 + q;         // 0..63
        int i = idx >> 3, j = idx & 7;
        out[((long)n * 64 + idx) * 2 + 0] = w[i * 16 + j];
        out[((long)n * 64 + idx) * 2 + 1] = w[(i + 8) * 16 + j];
    }
}

extern "C" void kernel_launch(void* const* d_in, const int* in_sizes, int n_in,
                              void* d_out, int out_size, void* d_ws, size_t ws_size,
                              hipStream_t stream) {
    const float* x    = (const float*)d_in[0];
    const float* A_re = (const float*)d_in[1];
    const float* A_im = (const float*)d_in[2];
    float* part = (float*)d_ws;  // NN*NCHUNK*256 floats = 2 MB

    dev_phase1<<<NN * NCHUNK / 8, 256, 0, stream>>>(x, A_re, A_im, part);
    dev_phase2<<<NN / 8, 256, 0, stream>>>(part, (float*)d_out);
}